// SAGEOneConvLayer_50448685859138
// MI455X (gfx1250) — compile-verified
//
#include <hip/hip_runtime.h>
#include <hip/hip_bf16.h>

// ---------------------------------------------------------------------------
// SAGE conv layer for MI455X (gfx1250, wave32, WMMA).
//   msg_sum = segment_sum(h[src], dst); deg = segment_sum(1, dst)
//   h_total = [h, msg_sum/max(deg,1)]            [N, 256]
//   out     = relu((h_total@W1 + b1)@W2 + b2)    [N, 128]
// ---------------------------------------------------------------------------

#define IN_DIM 128
#define TOT    256   // 2*IN_DIM
#define HID    256
#define OUTD   128
#define LDA    257   // LDS row stride (pad vs 64 banks)

typedef __attribute__((ext_vector_type(2))) float v2f;
typedef __attribute__((ext_vector_type(8))) float v8f;

// ---------------------------------------------------------------- zero ws
__global__ void sage_zero_kernel(float* __restrict__ p, long long n) {
    long long i = (long long)blockIdx.x * blockDim.x + threadIdx.x;
    long long stride = (long long)gridDim.x * blockDim.x;
    for (; i < n; i += stride) p[i] = 0.0f;
}

// ------------------------------------------------------- edge scatter-add
// one wave per edge; lane handles a 4-float feature chunk
__global__ __launch_bounds__(256) void sage_scatter_kernel(
    const float* __restrict__ h, const int* __restrict__ src,
    const int* __restrict__ dst, float* __restrict__ msg,
    float* __restrict__ deg, int E) {
    long long gid = (long long)blockIdx.x * blockDim.x + threadIdx.x;
    int e = (int)(gid >> 5);
    int chunk = (int)(gid & 31);
    if (e >= E) return;
    int s = src[e];
    int d = dst[e];
    const float4 v = *(const float4*)(h + (size_t)s * IN_DIM + chunk * 4);
    float* p = msg + (size_t)d * IN_DIM + chunk * 4;
    atomicAdd(p + 0, v.x);
    atomicAdd(p + 1, v.y);
    atomicAdd(p + 2, v.z);
    atomicAdd(p + 3, v.w);
    if (chunk == 0) atomicAdd(deg + d, 1.0f);
}

// --------------------------------------------------- fused 2-layer MLP
// 256 threads = 8 waves; 16 nodes (rows) per workgroup.
__global__ __launch_bounds__(256) void sage_mlp_kernel(
    const float* __restrict__ h, const float* __restrict__ msg,
    const float* __restrict__ deg,
    const float* __restrict__ W1, const float* __restrict__ b1,
    const float* __restrict__ W2, const float* __restrict__ b2,
    float* __restrict__ out, int N) {

    __shared__ float sh_in[16 * LDA];   // h_total tile  [16, 256]
    __shared__ float sh_h1[16 * LDA];   // hidden tile   [16, 256]
    __shared__ float sh_inv[16];

    const int tid  = threadIdx.x;
    const int lane = tid & 31;
    const int wave = tid >> 5;
    const int half = lane >> 4;   // 0 | 1
    const int lm   = lane & 15;   // lane % 16

    const int rowBase = blockIdx.x * 16;

    // ---- stage 0: inverse degrees, then load h_total tile into LDS ----
    if (tid < 16) {
        int node = rowBase + tid;
        float d = (node < N) ? deg[node] : 1.0f;
        sh_inv[tid] = 1.0f / fmaxf(d, 1.0f);
    }
    __syncthreads();

    for (int i = tid; i < 16 * TOT; i += 256) {
        int r = i >> 8;          // row in tile
        int c = i & 255;         // col 0..255
        int node = rowBase + r;
        float v = 0.0f;
        if (node < N) {
            v = (c < IN_DIM) ? h[(size_t)node * IN_DIM + c]
                             : msg[(size_t)node * IN_DIM + (c - IN_DIM)] * sh_inv[r];
        }
        sh_in[r * LDA + c] = v;
    }
    __syncthreads();

    // ---- stage 1: h1 = h_total @ W1 + b1  (16 N-tiles over 8 waves) ----
    for (int nt = wave; nt < HID / 16; nt += 8) {
        const int n0 = nt * 16;
        const float bb = b1[n0 + lm];
        v8f c;
#pragma unroll
        for (int r = 0; r < 8; ++r) c[r] = bb;

        const float* arow = &sh_in[lm * LDA];
#pragma unroll 4
        for (int k = 0; k < TOT; k += 4) {
            const int ka = k + 2 * half;                   // K for this half-wave
            v2f a = { arow[ka], arow[ka + 1] };
            v2f b = { W1[(size_t)ka * HID + n0 + lm],
                      W1[(size_t)(ka + 1) * HID + n0 + lm] };
            c = __builtin_amdgcn_wmma_f32_16x16x4_f32(
                    false, a, false, b, (short)0, c, false, false);
        }
        // C/D layout: vgpr r, lane L -> (M = r + 8*half, N = lm)
#pragma unroll
        for (int r = 0; r < 8; ++r)
            sh_h1[(r + 8 * half) * LDA + n0 + lm] = c[r];
    }
    __syncthreads();

    // ---- stage 2: out = relu(h1 @ W2 + b2)  (8 N-tiles, 1 per wave) ----
    {
        const int nt = wave;                  // OUTD/16 == 8 == #waves
        const int n0 = nt * 16;
        const float bb = b2[n0 + lm];
        v8f c;
#pragma unroll
        for (int r = 0; r < 8; ++r) c[r] = bb;

        const float* arow = &sh_h1[lm * LDA];
#pragma unroll 4
        for (int k = 0; k < HID; k += 4) {
            const int ka = k + 2 * half;
            v2f a = { arow[ka], arow[ka + 1] };
            v2f b = { W2[(size_t)ka * OUTD + n0 + lm],
                      W2[(size_t)(ka + 1) * OUTD + n0 + lm] };
            c = __builtin_amdgcn_wmma_f32_16x16x4_f32(
                    false, a, false, b, (short)0, c, false, false);
        }
#pragma unroll
        for (int r = 0; r < 8; ++r) {
            int node = rowBase + r + 8 * half;
            if (node < N)
                out[(size_t)node * OUTD + n0 + lm] = fmaxf(c[r], 0.0f);
        }
    }
}

// ---------------------------------------------------------------------------
extern "C" void kernel_launch(void* const* d_in, const int* in_sizes, int n_in,
                              void* d_out, int out_size, void* d_ws, size_t ws_size,
                              hipStream_t stream) {
    const float* h   = (const float*)d_in[0];
    const int*   src = (const int*)d_in[1];
    const int*   dst = (const int*)d_in[2];
    const float* W1  = (const float*)d_in[3];
    const float* b1  = (const float*)d_in[4];
    const float* W2  = (const float*)d_in[5];
    const float* b2  = (const float*)d_in[6];
    float* out = (float*)d_out;

    const int N = in_sizes[0] / IN_DIM;
    const int E = in_sizes[1];

    float* msg = (float*)d_ws;                 // [N, 128]
    float* deg = msg + (size_t)N * IN_DIM;     // [N]

    // zero accumulators
    long long zn = (long long)N * IN_DIM + N;
    sage_zero_kernel<<<2048, 256, 0, stream>>>(msg, zn);

    // edge scatter: one wave per edge
    long long sthreads = (long long)E * 32;
    int sblocks = (int)((sthreads + 255) / 256);
    sage_scatter_kernel<<<sblocks, 256, 0, stream>>>(h, src, dst, msg, deg, E);

    // fused MLP over 16-row tiles
    int mblocks = (N + 15) / 16;
    sage_mlp_kernel<<<mblocks, 256, 0, stream>>>(h, msg, deg, W1, b1, W2, b2, out, N);
}